// PINN_NS_12154757448258
// MI455X (gfx1250) — compile-verified
//
#include <hip/hip_runtime.h>

typedef float v2f __attribute__((ext_vector_type(2)));
typedef float v8f __attribute__((ext_vector_type(8)));

#define TILE_M 16
#define HID    256
#define STRIDE 268           // 256 + 12 pad: conflict-free banks, rows 8B-aligned
#define SSZ    (TILE_M * STRIDE)
#define NSTATE 6             // h, g_t, g_x, g_y, s_xx, s_yy
#define NU_C   (1.0f / 100.0f)
#define WPACK_LAYER (HID * HID)          // floats per packed hidden layer
#define WPACK_BYTES (5u * WPACK_LAYER * sizeof(float))

struct Params {
    const float* t; const float* x; const float* y;
    const float* W[7]; const float* b[7];
    const float* Wp;                     // packed hidden weights (or null)
    float* out; int n;
};

// Repack W1..W5 into fragment order: for K-chunk kg (4 rows), column n,
// packed[l][kg*1024 + n*4 + j] = W[l+1][(4*kg + j)*HID + n],  j = 0..3.
// A lane's B-fragment (2 consecutive K values) becomes one b64 load.
__global__ void pinn_ns_pack_kernel(const float* __restrict__ W1,
                                    const float* __restrict__ W2,
                                    const float* __restrict__ W3,
                                    const float* __restrict__ W4,
                                    const float* __restrict__ W5,
                                    float* __restrict__ Wp) {
    int idx = blockIdx.x * blockDim.x + threadIdx.x;   // 5 * 64 * 256 threads
    if (idx >= 5 * 64 * HID) return;
    int l   = idx / (64 * HID);
    int rem = idx % (64 * HID);
    int kg  = rem / HID;
    int n   = rem % HID;
    const float* W = (l == 0) ? W1 : (l == 1) ? W2 : (l == 2) ? W3
                   : (l == 3) ? W4 : W5;
    float* dst = Wp + l * WPACK_LAYER + kg * 1024 + n * 4;
    #pragma unroll
    for (int j = 0; j < 4; ++j)
        dst[j] = W[(4 * kg + j) * HID + n];            // 16B contiguous store
}

template <bool PACKED>
__launch_bounds__(256)
__global__ void pinn_ns_wmma_kernel(Params P) {
    extern __shared__ float lds[];
    float* S   = lds;                    // NSTATE * 16 * STRIDE state matrices
    float* txy = lds + NSTATE * SSZ;     // 48 floats: t[16], x[16], y[16]
    float* res = txy + 48;               // 16 * 18 final-layer results

    const int tid  = threadIdx.x;
    const int lane = tid & 31;
    const int wave = tid >> 5;
    const int lr   = lane & 15;          // row (A) or col (B/C) within 16
    const int half = (lane >> 4) & 1;    // 0: lanes 0-15, 1: lanes 16-31
    const int row0 = blockIdx.x * TILE_M;

    // ---- stage the 16 points' (t, x, y) ----
    if (tid < 48) {
        int m = tid & 15;
        int p = row0 + m; if (p >= P.n) p = P.n - 1;
        const float* src = (tid < 16) ? P.t : (tid < 32) ? P.x : P.y;
        txy[tid] = src[p];
    }
    __syncthreads();

    // ---- layer 0: 3 -> 256 (VALU; seeds jets) ----
    {
        const float* W0 = P.W[0]; const float* B0 = P.b[0];
        const int col = tid;                       // 256 threads == 256 cols
        const float w0 = W0[col], w1 = W0[HID + col], w2 = W0[2*HID + col];
        const float bb = B0[col];
        #pragma unroll 4
        for (int m = 0; m < TILE_M; ++m) {
            float z  = txy[m]*w0 + txy[16+m]*w1 + txy[32+m]*w2 + bb;
            float sz = __sinf(z), cz = __cosf(z);
            int base = m*STRIDE + col;
            S[0*SSZ + base] = sz;
            S[1*SSZ + base] = cz * w0;             // dh/dt
            S[2*SSZ + base] = cz * w1;             // dh/dx
            S[3*SSZ + base] = cz * w2;             // dh/dy
            S[4*SSZ + base] = -sz * w1 * w1;       // d2h/dx2
            S[5*SSZ + base] = -sz * w2 * w2;       // d2h/dy2
        }
    }
    __syncthreads();

    // ---- hidden layers 1..5: 256 -> 256 via fp32 WMMA ----
    const int c0 = (wave*2 + 0) * 16;
    const int c1 = (wave*2 + 1) * 16;
    for (int l = 1; l <= 5; ++l) {
        const float* Bl = P.b[l];

        v8f acc[NSTATE][2];
        #pragma unroll
        for (int s = 0; s < NSTATE; ++s) {
            acc[s][0] = (v8f){0.f,0.f,0.f,0.f,0.f,0.f,0.f,0.f};
            acc[s][1] = (v8f){0.f,0.f,0.f,0.f,0.f,0.f,0.f,0.f};
        }

        // Direct path: lane reads W[(kk + 2*half + j)*256 + ct*16 + lr]
        const float* wlane = P.W[l] + (2*half)*HID + lr;
        // Packed path: lane reads b64 at Wp[(kk/4)*1024 + col*4 + 2*half]
        const float* plane = P.Wp + (l - 1)*WPACK_LAYER + lr*4 + 2*half;
        // A-fragment base: lane holds state row lr, K slot kk + 2*half + {0,1}
        const float* arow = S + lr*STRIDE + 2*half;

        #pragma unroll 2
        for (int kk = 0; kk < HID; kk += 4) {
            v2f b0f, b1f;
            if (PACKED) {
                const float* pk = plane + (kk >> 2) * 1024;
                b0f = *(const v2f*)(pk + c0 * 4);
                b1f = *(const v2f*)(pk + c1 * 4);
            } else {
                b0f.x = wlane[(kk+0)*HID + c0];
                b0f.y = wlane[(kk+1)*HID + c0];
                b1f.x = wlane[(kk+0)*HID + c1];
                b1f.y = wlane[(kk+1)*HID + c1];
            }

            v2f af[NSTATE];
            #pragma unroll
            for (int s = 0; s < NSTATE; ++s)
                af[s] = *(const v2f*)(arow + s*SSZ + kk);

            #pragma unroll
            for (int s = 0; s < NSTATE; ++s) {
                acc[s][0] = __builtin_amdgcn_wmma_f32_16x16x4_f32(
                    false, af[s], false, b0f, (short)0, acc[s][0], false, false);
                acc[s][1] = __builtin_amdgcn_wmma_f32_16x16x4_f32(
                    false, af[s], false, b1f, (short)0, acc[s][1], false, false);
            }
        }

        __syncthreads();   // all waves done reading old state

        // activation + jet chain rule + writeback (C layout: VGPR r -> M, lanes -> N)
        #pragma unroll
        for (int t2 = 0; t2 < 2; ++t2) {
            const int col = (wave*2 + t2)*16 + lr;
            const float bb = Bl[col];
            #pragma unroll
            for (int r = 0; r < 8; ++r) {
                const int m = half*8 + r;
                float z   = acc[0][t2][r] + bb;
                float sz  = __sinf(z), cz = __cosf(z);
                float gt  = acc[1][t2][r];
                float gx  = acc[2][t2][r];
                float gy  = acc[3][t2][r];
                float hxx = acc[4][t2][r];
                float hyy = acc[5][t2][r];
                int base = m*STRIDE + col;
                S[0*SSZ + base] = sz;
                S[1*SSZ + base] = cz * gt;
                S[2*SSZ + base] = cz * gx;
                S[3*SSZ + base] = cz * gy;
                S[4*SSZ + base] = cz * hxx - sz * gx * gx;
                S[5*SSZ + base] = cz * hyy - sz * gy * gy;
            }
        }
        __syncthreads();
    }

    // ---- final layer: 256 -> 3, no activation (tiny; VALU reduction) ----
    {
        const float* W6 = P.W[6]; const float* B6 = P.b[6];
        for (int idx = tid; idx < TILE_M * NSTATE * 3; idx += 256) {
            int s   = idx / (TILE_M * 3);
            int rem = idx % (TILE_M * 3);
            int m   = rem / 3, col = rem % 3;
            const float* hrow = S + s*SSZ + m*STRIDE;
            float a = (s == 0) ? B6[col] : 0.0f;
            #pragma unroll 8
            for (int k = 0; k < HID; ++k)
                a += hrow[k] * W6[k*3 + col];
            res[m*18 + s*3 + col] = a;
        }
    }
    __syncthreads();

    // ---- assemble Navier-Stokes residuals ----
    if (tid < TILE_M) {
        int m = tid, p = row0 + m;
        if (p < P.n) {
            const float* r = res + m*18;
            float u = r[0],    v    = r[1];
            float u_t = r[3],  v_t  = r[4];
            float u_x = r[6],  v_x  = r[7],  p_x = r[8];
            float u_y = r[9],  v_y  = r[10], p_y = r[11];
            float u_xx = r[12], v_xx = r[13];
            float u_yy = r[15], v_yy = r[16];
            float eq1 = u_t + (u*u_x + v*u_y) + p_x - NU_C*(u_xx + u_yy);
            float eq2 = v_t + (u*v_x + v*v_y) + p_y - NU_C*(v_xx + v_yy);
            float eq3 = u_x + v_y;
            P.out[p]           = eq1;
            P.out[P.n + p]     = eq2;
            P.out[2*P.n + p]   = eq3;
        }
    }
}

extern "C" void kernel_launch(void* const* d_in, const int* in_sizes, int n_in,
                              void* d_out, int out_size, void* d_ws, size_t ws_size,
                              hipStream_t stream) {
    (void)n_in; (void)out_size;
    Params P;
    P.t = (const float*)d_in[0];
    P.x = (const float*)d_in[1];
    P.y = (const float*)d_in[2];
    for (int i = 0; i < 7; ++i) {
        P.W[i] = (const float*)d_in[3 + 2*i];
        P.b[i] = (const float*)d_in[4 + 2*i];
    }
    P.out = (float*)d_out;
    P.n   = in_sizes[0];

    const int nblocks = (P.n + TILE_M - 1) / TILE_M;
    const size_t shmem = (size_t)(NSTATE * SSZ + 48 + TILE_M * 18) * sizeof(float);
    const bool packed = (ws_size >= (size_t)WPACK_BYTES) && (d_ws != nullptr);

    if (packed) {
        float* Wp = (float*)d_ws;
        const int nrepack = 5 * 64 * HID;            // one thread per (l, kg, n)
        hipLaunchKernelGGL(pinn_ns_pack_kernel,
                           dim3((nrepack + 255) / 256), dim3(256), 0, stream,
                           P.W[1], P.W[2], P.W[3], P.W[4], P.W[5], Wp);
        P.Wp = Wp;
        hipLaunchKernelGGL(pinn_ns_wmma_kernel<true>,
                           dim3(nblocks), dim3(256), shmem, stream, P);
    } else {
        P.Wp = (const float*)d_ws;                   // unused on this path
        hipLaunchKernelGGL(pinn_ns_wmma_kernel<false>,
                           dim3(nblocks), dim3(256), shmem, stream, P);
    }
}